// TorchSVF_47132971106994
// MI455X (gfx1250) — compile-verified
//
#include <hip/hip_runtime.h>
#include <cstdint>
#include <cstddef>

// ---------------------------------------------------------------------------
// Time-varying SVF (lpf mode), T=32768, B=256.
// Strategy: linear-recurrence chunked parallel scan (3 passes) to break the
// serial T dependency; TDM (tensor_load_to_lds) stages input tiles into LDS
// in pass 3. Compute is trivial (~0.25 GFLOP); mandatory traffic 96 MB
// (~4 us at 23.3 TB/s); the real cost is the serial dependency chain, which
// the chunked scan spreads over 262K threads. Inputs (64 MB) fit in 192 MB L2
// so pass-3 rereads are L2 hits.
// ---------------------------------------------------------------------------

#define SR_F      44100.0f
#define PI_F      3.14159265358979323846f

constexpr int BATCH = 256;   // channels (fixed by reference)
constexpr int CHUNK = 32;    // time steps per chunk (tile = 32x256 f32 = 32KB)

typedef unsigned int u32x4 __attribute__((ext_vector_type(4)));
typedef int          i32x8 __attribute__((ext_vector_type(8)));
typedef int          i32x4 __attribute__((ext_vector_type(4)));

// ---------------------------------------------------------------------------
// TDM helper: DMA a rows x 256 f32 tile (row-major, row stride 256) from
// global memory into LDS. D# layout per CDNA5 ISA ch.8 (CDNA5 ISA).
// This toolchain exposes the 6-arg builtin:
//   (uint32x4 g0, int32x8 g1, int32x4 g2, int32x4 g3, int32x8 pad, i32 cpol)
// ---------------------------------------------------------------------------
#if defined(__gfx1250__) && __has_builtin(__builtin_amdgcn_tensor_load_to_lds)
#define USE_TDM 1
#else
#define USE_TDM 0
#endif

#if USE_TDM
__device__ __forceinline__ void tdm_load_tile_f32(const float* gsrc,
                                                  unsigned int lds_byte_off,
                                                  unsigned int rows,
                                                  unsigned int cols,
                                                  unsigned int tensor_rows)
{
    unsigned long long ga = (unsigned long long)(uintptr_t)gsrc;

    // --- D# group 0 (4 dwords) ---
    u32x4 g0;
    g0.x = 1u;                                   // count=1, user mode, no gather
    g0.y = lds_byte_off;                         // lds_addr [63:32]
    g0.z = (unsigned int)(ga & 0xFFFFFFFFu);     // global_addr [95:64]
    g0.w = (unsigned int)((ga >> 32) & 0x01FFFFFFu)  // global_addr [120:96]
         | 0x80000000u;                          // type=2 ("image") at [127:126]

    // --- D# group 1 (8 dwords) ---
    unsigned int dim0 = cols;            // fastest (contiguous) dim = batch
    unsigned int dim1 = tensor_rows;     // time dimension (OOB bound)
    unsigned int str0 = cols;            // row stride in elements
    i32x8 g1;
    g1[0] = (int)(2u << 16);                                   // data_size=4B; wg_mask=0
    g1[1] = (int)((dim0 & 0xFFFFu) << 16);                     // tensor_dim0 lo16
    g1[2] = (int)(((dim0 >> 16) & 0xFFFFu) |
                  ((dim1 & 0xFFFFu) << 16));                   // dim0 hi | dim1 lo
    g1[3] = (int)(((dim1 >> 16) & 0xFFFFu) |
                  ((cols & 0xFFFFu) << 16));                   // dim1 hi | tile_dim0
    g1[4] = (int)(rows & 0xFFFFu);                             // tile_dim1; tile_dim2=0
    g1[5] = (int)str0;                                         // tensor_dim0_stride lo32
    g1[6] = 0;                                                 // stride hi | dim1_stride lo (unused, 2D)
    g1[7] = 0;

    i32x4 z4 = {0, 0, 0, 0};                                   // groups 2/3 unused (2D)
    i32x8 z8 = {0, 0, 0, 0, 0, 0, 0, 0};
    __builtin_amdgcn_tensor_load_to_lds(g0, g1, z4, z4, z8, 0);
}
#endif

// ---------------------------------------------------------------------------
// Per-sample SVF coefficients. Affine state update:
//   h0' = (1 - 2gd r) h0 - 2gd h1        + 2gd  a
//   h1' = (2g - 2g^2 d r) h0 + (1-2g^2 d) h1 + 2g^2 d a
// with g = tan(pi*(fc + c*md)/SR), d = 1/(1 + rc*g + g^2), r = rc + g.
// ---------------------------------------------------------------------------

// Pass 1: per-(chunk, channel) composition of the chunk's affine maps.
__global__ void __launch_bounds__(BATCH)
svf_pass1(const float* __restrict__ audio, const float* __restrict__ ctrl,
          const float* __restrict__ p_cut, const float* __restrict__ p_res,
          const float* __restrict__ p_mod, float* __restrict__ ws, int C)
{
    const int b = threadIdx.x;
    const int c = blockIdx.x;
    const float fc = p_cut[0];
    const float rc = 1.0f / p_res[0];
    const float md = p_mod[0];
    const float w0 = PI_F / SR_F;

    float m00 = 1.f, m01 = 0.f, m10 = 0.f, m11 = 1.f, v0 = 0.f, v1 = 0.f;
    size_t base = (size_t)c * CHUNK * BATCH + b;
#pragma unroll 4
    for (int i = 0; i < CHUNK; ++i) {
        float a  = audio[base + (size_t)i * BATCH];
        float cc = ctrl [base + (size_t)i * BATCH];
        float g   = __tanf(w0 * (fc + cc * md));
        float d   = 1.0f / (1.0f + rc * g + g * g);
        float r   = rc + g;
        float tgd = 2.0f * g * d;
        float a00 = 1.0f - tgd * r;
        float a01 = -tgd;
        float a10 = 2.0f * g - g * tgd * r;
        float a11 = 1.0f - g * tgd;
        float b0  = tgd * a;
        float b1  = g * tgd * a;
        float n00 = a00 * m00 + a01 * m10;
        float n01 = a00 * m01 + a01 * m11;
        float n10 = a10 * m00 + a11 * m10;
        float n11 = a10 * m01 + a11 * m11;
        float nv0 = a00 * v0 + a01 * v1 + b0;
        float nv1 = a10 * v0 + a11 * v1 + b1;
        m00 = n00; m01 = n01; m10 = n10; m11 = n11; v0 = nv0; v1 = nv1;
    }
    size_t CB  = (size_t)C * BATCH;
    size_t idx = (size_t)c * BATCH + b;
    ws[0 * CB + idx] = m00; ws[1 * CB + idx] = m01;
    ws[2 * CB + idx] = m10; ws[3 * CB + idx] = m11;
    ws[4 * CB + idx] = v0;  ws[5 * CB + idx] = v1;
}

// Pass 2: sequential scan over chunks (tiny: C iterations of a 2x2 affine).
__global__ void __launch_bounds__(BATCH)
svf_pass2(float* __restrict__ ws, int C)
{
    const int b = threadIdx.x;
    size_t CB = (size_t)C * BATCH;
    float h0 = 0.f, h1 = 0.f;
    for (int c = 0; c < C; ++c) {
        size_t idx = (size_t)c * BATCH + b;
        ws[6 * CB + idx] = h0;
        ws[7 * CB + idx] = h1;
        float n0 = ws[0 * CB + idx] * h0 + ws[1 * CB + idx] * h1 + ws[4 * CB + idx];
        float n1 = ws[2 * CB + idx] * h0 + ws[3 * CB + idx] * h1 + ws[5 * CB + idx];
        h0 = n0; h1 = n1;
    }
}

// Pass 3: re-run the exact reference recurrence per chunk; inputs staged into
// LDS via the Tensor Data Mover (one DMA per array, issued by wave 0).
__global__ void __launch_bounds__(BATCH)
svf_pass3(const float* __restrict__ audio, const float* __restrict__ ctrl,
          const float* __restrict__ p_cut, const float* __restrict__ p_res,
          const float* __restrict__ p_mod, const float* __restrict__ ws,
          float* __restrict__ out, int C, int T)
{
    __shared__ float sA[CHUNK * BATCH];
    __shared__ float sC[CHUNK * BATCH];

    const int b = threadIdx.x;
    const int c = blockIdx.x;
    const size_t tile_base = (size_t)c * CHUNK * BATCH;

#if USE_TDM
    if (threadIdx.x == 0) {
        // Flat LDS addresses carry the LDS byte offset in the low 32 bits.
        unsigned int offA = (unsigned int)(uintptr_t)&sA[0];
        unsigned int offC = (unsigned int)(uintptr_t)&sC[0];
        tdm_load_tile_f32(audio + tile_base, offA, CHUNK, BATCH, (unsigned)T);
        tdm_load_tile_f32(ctrl  + tile_base, offC, CHUNK, BATCH, (unsigned)T);
        __builtin_amdgcn_s_wait_tensorcnt(0);
    }
    __syncthreads();
#else
    for (int i = threadIdx.x; i < CHUNK * BATCH; i += blockDim.x) {
        sA[i] = audio[tile_base + i];
        sC[i] = ctrl [tile_base + i];
    }
    __syncthreads();
#endif

    const float fc = p_cut[0];
    const float rc = 1.0f / p_res[0];
    const float md = p_mod[0];
    const float w0 = PI_F / SR_F;

    size_t CB = (size_t)C * BATCH;
    float h0 = ws[6 * CB + (size_t)c * BATCH + b];
    float h1 = ws[7 * CB + (size_t)c * BATCH + b];

#pragma unroll 4
    for (int i = 0; i < CHUNK; ++i) {
        float a  = sA[i * BATCH + b];   // bank-conflict-free (consecutive dwords)
        float cc = sC[i * BATCH + b];
        float g   = __tanf(w0 * (fc + cc * md));
        float d   = 1.0f / (1.0f + rc * g + g * g);
        float rho = rc + g;
        float hpf = d * (a - rho * h0 - h1);
        float bpf = g * hpf + h0;
        float lpf = g * bpf + h1;
        h0 = g * hpf + bpf;
        h1 = g * bpf + lpf;
        out[tile_base + (size_t)i * BATCH + b] = lpf;  // MODE == "lpf"
    }
}

// Fallback (only if workspace is too small): one thread per channel, full scan.
__global__ void __launch_bounds__(BATCH)
svf_naive(const float* __restrict__ audio, const float* __restrict__ ctrl,
          const float* __restrict__ p_cut, const float* __restrict__ p_res,
          const float* __restrict__ p_mod, float* __restrict__ out, int T)
{
    const int b = threadIdx.x;
    const float fc = p_cut[0];
    const float rc = 1.0f / p_res[0];
    const float md = p_mod[0];
    const float w0 = PI_F / SR_F;
    float h0 = 0.f, h1 = 0.f;
    for (int t = 0; t < T; ++t) {
        size_t idx = (size_t)t * BATCH + b;
        float a  = audio[idx];
        float cc = ctrl [idx];
        float g   = __tanf(w0 * (fc + cc * md));
        float d   = 1.0f / (1.0f + rc * g + g * g);
        float rho = rc + g;
        float hpf = d * (a - rho * h0 - h1);
        float bpf = g * hpf + h0;
        float lpf = g * bpf + h1;
        h0 = g * hpf + bpf;
        h1 = g * bpf + lpf;
        out[idx] = lpf;
    }
}

extern "C" void kernel_launch(void* const* d_in, const int* in_sizes, int n_in,
                              void* d_out, int out_size, void* d_ws, size_t ws_size,
                              hipStream_t stream)
{
    const float* audio = (const float*)d_in[0];
    const float* ctrl  = (const float*)d_in[1];
    const float* cut   = (const float*)d_in[2];
    const float* res   = (const float*)d_in[3];
    const float* mod   = (const float*)d_in[4];
    float* out = (float*)d_out;

    const int T = in_sizes[0] / BATCH;   // 32768
    const int C = T / CHUNK;             // 1024 chunks

    const size_t ws_need = (size_t)8 * C * BATCH * sizeof(float);  // 8 MB
    if ((size_t)ws_size < ws_need || (T % CHUNK) != 0) {
        svf_naive<<<1, BATCH, 0, stream>>>(audio, ctrl, cut, res, mod, out, T);
        return;
    }

    float* ws = (float*)d_ws;
    svf_pass1<<<C, BATCH, 0, stream>>>(audio, ctrl, cut, res, mod, ws, C);
    svf_pass2<<<1, BATCH, 0, stream>>>(ws, C);
    svf_pass3<<<C, BATCH, 0, stream>>>(audio, ctrl, cut, res, mod, ws, out, C, T);
}